// GraphConvLSTMCell_18614388261510
// MI455X (gfx1250) — compile-verified
//
#include <hip/hip_runtime.h>
#include <math.h>

typedef float v2f __attribute__((ext_vector_type(2)));
typedef float v8f __attribute__((ext_vector_type(8)));

#define FDIM 64
#define HDIM 32
#define KDIM 96    // F + H
#define GDIM 128   // 4 * H

// ---------------------------------------------------------------------------
// GEMM: xw[BN,128] = [x | h] @ W  using V_WMMA_F32_16X16X4_F32.
// One block (256 thr = 8 waves) computes a 16-row x 128-col tile.
// Wave w handles columns [16w, 16w+16). K=96 -> 24 WMMA steps.
// ---------------------------------------------------------------------------
__global__ __launch_bounds__(256) void gemm_wmma_f32_kernel(
    const float* __restrict__ x, const float* __restrict__ h,
    const float* __restrict__ W, float* __restrict__ xw, int N) {
  __shared__ float Atile[16 * KDIM];     // 6 KB
  __shared__ float Wlds[KDIM * GDIM];    // 48 KB

  const int tid = threadIdx.x;
  const int m0  = blockIdx.x * 16;       // BN divisible by 16

  // Stage full W (96x128) in LDS (L2-resident anyway, but keeps inner loop on DS)
  for (int i = tid; i < KDIM * GDIM; i += 256) Wlds[i] = W[i];
  // Stage A tile: combined = concat(x, h) along feature dim
  for (int i = tid; i < 16 * KDIM; i += 256) {
    int r = i / KDIM, c = i % KDIM;
    int row = m0 + r;
    int bb = row / N, node = row - bb * N;
    Atile[i] = (c < FDIM) ? x[((long long)bb * N + node) * FDIM + c]
                          : h[((long long)bb * N + node) * HDIM + (c - FDIM)];
  }
  __syncthreads();

  const int wave = tid >> 5;     // 0..7 -> N tile
  const int lane = tid & 31;
  const int half = lane >> 4;    // 0/1
  const int lr   = lane & 15;    // A-row / B,C-col within tile
  const int n0   = wave * 16;

  v8f acc = {};
#pragma unroll
  for (int k0 = 0; k0 < KDIM; k0 += 4) {
    const int ka = k0 + 2 * half;
    v2f a, b;
    a.x = Atile[lr * KDIM + ka];
    a.y = Atile[lr * KDIM + ka + 1];
    b.x = Wlds[ka * GDIM + n0 + lr];
    b.y = Wlds[(ka + 1) * GDIM + n0 + lr];
    acc = __builtin_amdgcn_wmma_f32_16x16x4_f32(
        false, a, false, b, (short)0, acc, false, false);
  }

#pragma unroll
  for (int i = 0; i < 8; ++i) {
    const int row = m0 + i + 8 * half;
    xw[(long long)row * GDIM + n0 + lr] = acc[i];
  }
}

// ---------------------------------------------------------------------------
// Degree over dst indices (self-loops folded in analytically later).
// ---------------------------------------------------------------------------
__global__ void degree_kernel(const int* __restrict__ dst,
                              float* __restrict__ deg, int E) {
  int e = blockIdx.x * blockDim.x + threadIdx.x;
  if (e < E) atomicAdd(&deg[dst[e]], 1.0f);
}

__global__ void dinv_kernel(float* __restrict__ deg, int N) {
  int n = blockIdx.x * blockDim.x + threadIdx.x;
  if (n < N) deg[n] = rsqrtf(deg[n] + 1.0f);  // +1 accounts for self-loop
}

// ---------------------------------------------------------------------------
// Gather-scale-scatter: one wave (32 lanes) per edge, lane = 4-channel group.
// Edges [0,E) are graph edges; [E, E+N) are self-loops.
// ---------------------------------------------------------------------------
__global__ __launch_bounds__(256) void scatter_kernel(
    const int* __restrict__ src, const int* __restrict__ dst,
    const float* __restrict__ dinv, const float* __restrict__ xw,
    float* __restrict__ agg, int E, int N) {
  long long t = (long long)blockIdx.x * blockDim.x + threadIdx.x;
  int lane = (int)(t & 31);
  long long e = t >> 5;
  if (e >= (long long)E + N) return;

  int s, d;
  if (e < E) { s = src[e]; d = dst[e]; }
  else       { s = d = (int)(e - E); }
  const float norm = dinv[s] * dinv[d];
  const int c = lane * 4;

#pragma unroll
  for (int bb = 0; bb < 2; ++bb) {
    const float4 v = *(const float4*)(xw + ((long long)bb * N + s) * GDIM + c);
    float* base = agg + ((long long)bb * N + d) * GDIM + c;
    atomicAdd(base + 0, v.x * norm);
    atomicAdd(base + 1, v.y * norm);
    atomicAdd(base + 2, v.z * norm);
    atomicAdd(base + 3, v.w * norm);
  }
}

// ---------------------------------------------------------------------------
// LSTM gating: one thread per (b, n, h). out = [h_next | c_next] flat.
// ---------------------------------------------------------------------------
__global__ void lstm_kernel(const float* __restrict__ agg,
                            const float* __restrict__ bias,
                            const float* __restrict__ c_cur,
                            float* __restrict__ out, int BN) {
  int t = blockIdx.x * blockDim.x + threadIdx.x;
  if (t >= BN * HDIM) return;
  const int bn = t / HDIM, hc = t - bn * HDIM;
  const float* row = agg + (long long)bn * GDIM;

  const float ci = row[hc]      + bias[hc];
  const float cf = row[hc + 32] + bias[hc + 32];
  const float co = row[hc + 64] + bias[hc + 64];
  const float cg = row[hc + 96] + bias[hc + 96];

  const float ig = 1.0f / (1.0f + __expf(-ci));
  const float fg = 1.0f / (1.0f + __expf(-cf));
  const float og = 1.0f / (1.0f + __expf(-co));
  const float g  = tanhf(cg);

  const float cn = fg * c_cur[t] + ig * g;
  const float hn = og * tanhf(cn);
  out[t] = hn;
  out[(long long)BN * HDIM + t] = cn;
}

// ---------------------------------------------------------------------------
extern "C" void kernel_launch(void* const* d_in, const int* in_sizes, int n_in,
                              void* d_out, int out_size, void* d_ws, size_t ws_size,
                              hipStream_t stream) {
  const float* x     = (const float*)d_in[0];
  const int*   edges = (const int*)d_in[1];   // [2, E]: row0=src, row1=dst
  const float* h_cur = (const float*)d_in[2];
  const float* c_cur = (const float*)d_in[3];
  const float* W     = (const float*)d_in[4];
  const float* bias  = (const float*)d_in[5];
  float*       out   = (float*)d_out;

  const int E  = in_sizes[1] / 2;           // 800000
  const int BN = in_sizes[2] / HDIM;        // B*N = 100000
  const int N  = BN / 2;                    // B = 2
  const int* src = edges;
  const int* dst = edges + E;

  float* xw  = (float*)d_ws;                // BN*128 floats
  float* agg = xw + (size_t)BN * GDIM;      // BN*128 floats
  float* deg = agg + (size_t)BN * GDIM;     // N floats (deg -> dinv in place)

  hipMemsetAsync(agg, 0, (size_t)BN * GDIM * sizeof(float), stream);
  hipMemsetAsync(deg, 0, (size_t)N * sizeof(float), stream);

  // 1) GEMM via fp32 WMMA
  gemm_wmma_f32_kernel<<<BN / 16, 256, 0, stream>>>(x, h_cur, W, xw, N);

  // 2) degrees + rsqrt normalization
  degree_kernel<<<(E + 255) / 256, 256, 0, stream>>>(dst, deg, E);
  dinv_kernel<<<(N + 255) / 256, 256, 0, stream>>>(deg, N);

  // 3) edge aggregation (incl. self-loops), 32 lanes per edge
  long long sthreads = ((long long)E + N) * 32;
  unsigned sblocks = (unsigned)((sthreads + 255) / 256);
  scatter_kernel<<<sblocks, 256, 0, stream>>>(src, dst, deg, xw, agg, E, N);

  // 4) LSTM gating -> d_out = [h_next | c_next]
  lstm_kernel<<<(BN * HDIM + 255) / 256, 256, 0, stream>>>(agg, bias, c_cur, out, BN);
}